// GATWithGlobal_v2_19920058319254
// MI455X (gfx1250) — compile-verified
//
#include <hip/hip_runtime.h>
#include <hip/hip_bf16.h>
#include <limits.h>

typedef float v2f __attribute__((ext_vector_type(2)));
typedef float v8f __attribute__((ext_vector_type(8)));

// ---------------------------------------------------------------------------
// helpers: order-preserving float<->int for atomicMax on floats
// ---------------------------------------------------------------------------
__device__ __forceinline__ int f32_ord(float f) {
    int i = __float_as_int(f);
    return (i < 0) ? (i ^ 0x7fffffff) : i;
}
__device__ __forceinline__ float ord_f32(int i) {
    return __int_as_float((i < 0) ? (i ^ 0x7fffffff) : i);
}

// ---------------------------------------------------------------------------
// WMMA GEMM:  D[M, Nout] = X[M, K] @ W[Nout, K]^T     (f32, 16x16x4 WMMA)
// grid.x = M/16 tiles, grid.y covers Nout in chunks of 64 (4 waves per block,
// one 16-wide N-tile per wave).  Assumes M%16==0, Nout%16==0, K%8==0.
// A-layout per lane (hi=lane>>4, lo=lane&15): row m0+lo, cols {k+2hi, k+2hi+1}
// B-layout per lane:                          col n0+lo, rows {k+2hi, k+2hi+1}
// D-layout: vgpr j -> row m0 + j + 8*hi, col n0+lo
// ---------------------------------------------------------------------------
template <int K>
__global__ __launch_bounds__(128)
void wmma_gemm_xwt(const float* __restrict__ X, const float* __restrict__ W,
                   float* __restrict__ D, int Nout) {
    const int lane = threadIdx.x & 31;
    const int wv   = threadIdx.x >> 5;
    const int hi   = lane >> 4;
    const int lo   = lane & 15;
    const int m0   = blockIdx.x * 16;
    const int n0   = (blockIdx.y * 4 + wv) * 16;

    const float* xrow = X + (size_t)(m0 + lo) * K + 2 * hi;
    const float* wrow = W + (size_t)(n0 + lo) * K + 2 * hi;

    v8f acc0 = {0.f, 0.f, 0.f, 0.f, 0.f, 0.f, 0.f, 0.f};
    v8f acc1 = {0.f, 0.f, 0.f, 0.f, 0.f, 0.f, 0.f, 0.f};
#pragma unroll
    for (int k = 0; k < K; k += 8) {
        v2f a0 = *(const v2f*)(xrow + k);
        v2f b0 = *(const v2f*)(wrow + k);
        v2f a1 = *(const v2f*)(xrow + k + 4);
        v2f b1 = *(const v2f*)(wrow + k + 4);
        acc0 = __builtin_amdgcn_wmma_f32_16x16x4_f32(false, a0, false, b0,
                                                     (short)0, acc0, false, false);
        acc1 = __builtin_amdgcn_wmma_f32_16x16x4_f32(false, a1, false, b1,
                                                     (short)0, acc1, false, false);
    }
    v8f acc = acc0 + acc1;

    float* drow = D + (size_t)(m0 + 8 * hi) * Nout + n0 + lo;
#pragma unroll
    for (int j = 0; j < 8; ++j) drow[(size_t)j * Nout] = acc[j];
}

// ---------------------------------------------------------------------------
// per-node attention coefficients: a_s[n,h] = sum_c h[n,h,c]*att_s[h,c], same
// for a_d.  One thread per (n,h); each thread reads a contiguous 256B row.
// ---------------------------------------------------------------------------
__global__ void att_coef(const float* __restrict__ h,
                         const float* __restrict__ att_s,
                         const float* __restrict__ att_d,
                         float* __restrict__ a_s, float* __restrict__ a_d,
                         int N, int H, int C) {
    int i = blockIdx.x * blockDim.x + threadIdx.x;
    if (i >= N * H) return;
    int hh = i % H;
    int n  = i / H;
    const float* row = h + (size_t)n * H * C + hh * C;
    const float* as  = att_s + hh * C;
    const float* ad  = att_d + hh * C;
    float s = 0.f, d = 0.f;
    for (int c = 0; c < C; ++c) {
        float v = row[c];
        s += v * as[c];
        d += v * ad[c];
    }
    a_s[i] = s;
    a_d[i] = d;
}

// ---------------------------------------------------------------------------
// fill int buffer (used for -inf init of segment-max buffers)
// ---------------------------------------------------------------------------
__global__ void fill_i32(int* __restrict__ p, int v, int n) {
    int i = blockIdx.x * blockDim.x + threadIdx.x;
    if (i < n) p[i] = v;
}

// ---------------------------------------------------------------------------
// edge pass 1: e = leaky_relu(a_s[src]+a_d[dst]); segment max via atomicMax.
// Edges [0,E) come from edge_index, [E, E+N) are self-loops.
// ---------------------------------------------------------------------------
__global__ void edge_max(const int* __restrict__ srcE, const int* __restrict__ dstE,
                         int E, int N,
                         const float* __restrict__ a_s, const float* __restrict__ a_d,
                         int H, int* __restrict__ m) {
    int e = blockIdx.x * blockDim.x + threadIdx.x;
    if (e >= E + N) return;
    int s = (e < E) ? srcE[e] : (e - E);
    int d = (e < E) ? dstE[e] : (e - E);
    for (int h = 0; h < H; ++h) {
        float v = a_s[s * H + h] + a_d[d * H + h];
        v = (v > 0.f) ? v : 0.2f * v;
        atomicMax(&m[d * H + h], f32_ord(v));
    }
}

// ---------------------------------------------------------------------------
// edge pass 2: one WAVE per edge.  p = exp(e - m[dst]);  ssum[dst] += p;
// agg[dst, f] += p * h[src, f] for all F = H*C features (lanes cover F).
// Normalization by ssum happens later, which is equivalent to alpha-weighting.
// Head index for feature f = lane + 32*j is (32*j)/C (compile-time, lane<32).
// ---------------------------------------------------------------------------
template <int H, int C>
__global__ __launch_bounds__(256)
void edge_scatter(const int* __restrict__ srcE, const int* __restrict__ dstE,
                  int E, int N,
                  const float* __restrict__ a_s, const float* __restrict__ a_d,
                  const int* __restrict__ m, const float* __restrict__ h,
                  float* __restrict__ ssum, float* __restrict__ agg) {
    constexpr int F = H * C;
    const int lane = threadIdx.x & 31;
    const int wid  = (blockIdx.x * blockDim.x + threadIdx.x) >> 5;
    if (wid >= E + N) return;
    int s = (wid < E) ? srcE[wid] : (wid - E);
    int d = (wid < E) ? dstE[wid] : (wid - E);

    const float* hs = h + (size_t)s * F;
    float* ad       = agg + (size_t)d * F;
    __builtin_prefetch(hs, 0, 0);

    // every lane computes all H edge probabilities (H<=4; loads are L0/L2 hits)
    float ph[H];
#pragma unroll
    for (int h0 = 0; h0 < H; ++h0) {
        float v = a_s[s * H + h0] + a_d[d * H + h0];
        v       = (v > 0.f) ? v : 0.2f * v;
        ph[h0]  = __expf(v - ord_f32(m[d * H + h0]));
    }
    // lanes 0..H-1 accumulate the softmax denominator exactly once per edge
    if (lane < H) {
        float v = a_s[s * H + lane] + a_d[d * H + lane];
        v       = (v > 0.f) ? v : 0.2f * v;
        atomicAdd(&ssum[d * H + lane], __expf(v - ord_f32(m[d * H + lane])));
    }
    // scatter weighted message: coalesced across lanes, atomics into L2
#pragma unroll
    for (int j = 0; j < F / 32; ++j) {
        int f = lane + 32 * j;
        atomicAdd(&ad[f], ph[(32 * j) / C] * hs[f]);
    }
}

// ---------------------------------------------------------------------------
// per-node epilogue: out = agg/max(ssum,1e-16) + bias, then ELU.
// (For layer 2, H=1 so concat=False head-mean is the identity.)
// ---------------------------------------------------------------------------
__global__ void normalize_elu(const float* __restrict__ agg,
                              const float* __restrict__ ssum,
                              const float* __restrict__ bias,
                              float* __restrict__ out, int N, int H, int C) {
    int F = H * C;
    int i = blockIdx.x * blockDim.x + threadIdx.x;
    if (i >= N * F) return;
    int f = i % F;
    int n = i / F;
    float v = agg[i] / fmaxf(ssum[n * H + f / C], 1e-16f) + bias[f];
    out[i]  = (v > 0.f) ? v : (__expf(v) - 1.f);  // ELU
}

// ---------------------------------------------------------------------------
// global mean pool (sums + counts; divide in head kernel)
// ---------------------------------------------------------------------------
__global__ void pool_sum(const float* __restrict__ h2, const int* __restrict__ batch,
                         int N, float* __restrict__ pooled, float* __restrict__ cnt) {
    int i = blockIdx.x * blockDim.x + threadIdx.x;
    if (i >= N * 64) return;
    int n = i >> 6;
    int f = i & 63;
    int b = batch[n];
    atomicAdd(&pooled[b * 64 + f], h2[i]);
    if (f == 0) atomicAdd(&cnt[b], 1.0f);
}

// ---------------------------------------------------------------------------
// head: pooled mean | global MLP (32->64->64, relu) | concat | fc 128->16
// one block of 64 threads per graph
// ---------------------------------------------------------------------------
__global__ __launch_bounds__(64)
void head_kernel(const float* __restrict__ pooled, const float* __restrict__ cnt,
                 const float* __restrict__ gf,
                 const float* __restrict__ gw1, const float* __restrict__ gb1,
                 const float* __restrict__ gw2, const float* __restrict__ gb2,
                 const float* __restrict__ fw, const float* __restrict__ fb,
                 float* __restrict__ out) {
    __shared__ float z[128];
    __shared__ float g1s[64];
    int b = blockIdx.x;
    int t = threadIdx.x;  // 0..63

    float c = fmaxf(cnt[b], 1.0f);
    z[t]    = pooled[b * 64 + t] / c;

    float acc = gb1[t];
    for (int k = 0; k < 32; ++k) acc += gf[b * 32 + k] * gw1[t * 32 + k];
    g1s[t] = fmaxf(acc, 0.f);
    __syncthreads();

    float acc2 = gb2[t];
    for (int k = 0; k < 64; ++k) acc2 += g1s[k] * gw2[t * 64 + k];
    z[64 + t] = fmaxf(acc2, 0.f);
    __syncthreads();

    if (t < 16) {
        float o = fb[t];
        for (int k = 0; k < 128; ++k) o += z[k] * fw[t * 128 + k];
        out[b * 16 + t] = o;
    }
}

// ---------------------------------------------------------------------------
// launcher
// ---------------------------------------------------------------------------
extern "C" void kernel_launch(void* const* d_in, const int* in_sizes, int n_in,
                              void* d_out, int out_size, void* d_ws, size_t ws_size,
                              hipStream_t stream) {
    (void)n_in; (void)out_size; (void)ws_size;
    const float* x   = (const float*)d_in[0];
    const int*   ei  = (const int*)d_in[1];
    const int*   bat = (const int*)d_in[2];
    const float* gf  = (const float*)d_in[3];
    const float* W1  = (const float*)d_in[4];
    const float* a1s = (const float*)d_in[5];
    const float* a1d = (const float*)d_in[6];
    const float* b1  = (const float*)d_in[7];
    const float* W2  = (const float*)d_in[8];
    const float* a2s = (const float*)d_in[9];
    const float* a2d = (const float*)d_in[10];
    const float* b2  = (const float*)d_in[11];
    const float* gw1 = (const float*)d_in[12];
    const float* gb1 = (const float*)d_in[13];
    const float* gw2 = (const float*)d_in[14];
    const float* gb2 = (const float*)d_in[15];
    const float* fw  = (const float*)d_in[16];
    const float* fb  = (const float*)d_in[17];
    float* out = (float*)d_out;

    const int Fin = 128, F1 = 256, H1 = 4, C = 64, F2 = 64, Gin = 32;
    const int N  = in_sizes[0] / Fin;   // 50000
    const int E  = in_sizes[1] / 2;     // 800000
    const int B  = in_sizes[3] / Gin;   // 64
    const int ET = E + N;               // edges incl. self-loops
    const int* srcE = ei;               // edge_index[0]
    const int* dstE = ei + E;           // edge_index[1]

    // ---- workspace carve ----
    float* h1   = (float*)d_ws;                 // [N, 256]  gemm1 out / elu out
    float* agg1 = h1 + (size_t)N * F1;          // [N, 256]
    float* as1  = agg1 + (size_t)N * F1;        // [N, 4]
    float* ad1  = as1 + (size_t)N * H1;         // [N, 4]
    float* s1   = ad1 + (size_t)N * H1;         // [N, 4]
    int*   m1   = (int*)(s1 + (size_t)N * H1);  // [N, 4]
    float* h2   = (float*)(m1 + (size_t)N * H1);// [N, 64]
    float* agg2 = h2 + (size_t)N * F2;          // [N, 64]
    float* as2  = agg2 + (size_t)N * F2;        // [N]
    float* ad2  = as2 + N;                      // [N]
    float* s2   = ad2 + N;                      // [N]
    int*   m2   = (int*)(s2 + N);               // [N]
    float* pooled = (float*)(m2 + N);           // [B, 64]
    float* cnt  = pooled + (size_t)B * F2;      // [B]

    const int TB = 256;

    // ================= GAT layer 1 =================
    // h1 = x @ W1^T  (WMMA f32)
    {
        dim3 grid(N / 16, F1 / 64);
        wmma_gemm_xwt<128><<<grid, 128, 0, stream>>>(x, W1, h1, F1);
    }
    att_coef<<<(N * H1 + TB - 1) / TB, TB, 0, stream>>>(h1, a1s, a1d, as1, ad1, N, H1, C);

    hipMemsetAsync(agg1, 0, (size_t)N * F1 * sizeof(float), stream);
    hipMemsetAsync(s1, 0, (size_t)N * H1 * sizeof(float), stream);
    fill_i32<<<(N * H1 + TB - 1) / TB, TB, 0, stream>>>(m1, INT_MIN, N * H1);

    edge_max<<<(ET + TB - 1) / TB, TB, 0, stream>>>(srcE, dstE, E, N, as1, ad1, H1, m1);
    edge_scatter<4, 64><<<(ET + 7) / 8, TB, 0, stream>>>(srcE, dstE, E, N,
                                                         as1, ad1, m1, h1, s1, agg1);
    normalize_elu<<<((size_t)N * F1 + TB - 1) / TB, TB, 0, stream>>>(agg1, s1, b1, h1,
                                                                     N, H1, C);

    // ================= GAT layer 2 =================
    // h2 = h1 @ W2^T  (WMMA f32)
    {
        dim3 grid(N / 16, F2 / 64);
        wmma_gemm_xwt<256><<<grid, 128, 0, stream>>>(h1, W2, h2, F2);
    }
    att_coef<<<(N + TB - 1) / TB, TB, 0, stream>>>(h2, a2s, a2d, as2, ad2, N, 1, C);

    hipMemsetAsync(agg2, 0, (size_t)N * F2 * sizeof(float), stream);
    hipMemsetAsync(s2, 0, (size_t)N * sizeof(float), stream);
    fill_i32<<<(N + TB - 1) / TB, TB, 0, stream>>>(m2, INT_MIN, N);

    edge_max<<<(ET + TB - 1) / TB, TB, 0, stream>>>(srcE, dstE, E, N, as2, ad2, 1, m2);
    edge_scatter<1, 64><<<(ET + 7) / 8, TB, 0, stream>>>(srcE, dstE, E, N,
                                                         as2, ad2, m2, h2, s2, agg2);
    normalize_elu<<<((size_t)N * F2 + TB - 1) / TB, TB, 0, stream>>>(agg2, s2, b2, h2,
                                                                     N, 1, C);

    // ================= pool + head =================
    hipMemsetAsync(pooled, 0, (size_t)B * F2 * sizeof(float), stream);
    hipMemsetAsync(cnt, 0, (size_t)B * sizeof(float), stream);
    pool_sum<<<((size_t)N * F2 + TB - 1) / TB, TB, 0, stream>>>(h2, bat, N, pooled, cnt);

    head_kernel<<<B, 64, 0, stream>>>(pooled, cnt, gf, gw1, gb1, gw2, gb2, fw, fb, out);
}